// SelfAttentionWeight_64123861729623
// MI455X (gfx1250) — compile-verified
//
#include <hip/hip_runtime.h>
#include <math.h>

typedef float v2f __attribute__((ext_vector_type(2)));
typedef float v8f __attribute__((ext_vector_type(8)));
typedef unsigned int v4u __attribute__((ext_vector_type(4)));
typedef int v4i __attribute__((ext_vector_type(4)));
typedef int v8i __attribute__((ext_vector_type(8)));

#define TT     64            // timesteps per LDS tile
#define ROWF   260           // padded LDS row length in floats (256 + 4 pad)
#define TILES  8             // tiles per block -> 512 t per block
#define TCHUNK (TT * TILES)
#define DDIM   256
#define TLEN   4096
#define BATCH  64
#define EPS_F  1e-7f

#if defined(__has_builtin)
#if __has_builtin(__builtin_amdgcn_tensor_load_to_lds)
#define HAVE_TDM 1
#endif
#endif

__device__ __forceinline__ float fast_tanh(float z) {
#if defined(__has_builtin) && __has_builtin(__builtin_amdgcn_tanhf)
  return __builtin_amdgcn_tanhf(z);   // native V_TANH_F32
#else
  return tanhf(z);
#endif
}

__device__ __forceinline__ float fast_exp(float z) {
#if defined(__has_builtin) && __has_builtin(__builtin_amdgcn_exp2f)
  return __builtin_amdgcn_exp2f(z * 1.44269504088896340736f);  // V_EXP_F32
#else
  return __expf(z);
#endif
}

#ifdef HAVE_TDM
// One TDM descriptor moves a 64x256 f32 tile (row-major, 1024B rows) from
// global memory into LDS, inserting 16B of padding after every 1024B row
// (pad_interval code 7 = 256 dwords, pad_amount code 3 = 4 dwords) so LDS
// rows land on a 260-float stride (bank-conflict-friendly).
__device__ __forceinline__ void tdm_load_tile(const float* gsrc, unsigned lds_off) {
  unsigned long long ga = (unsigned long long)(const void*)gsrc;
  v4u g0;
  g0[0] = 1u;                                   // count=1, user mode, no gather
  g0[1] = lds_off;                              // lds_addr [63:32]
  g0[2] = (unsigned)(ga & 0xFFFFFFFFu);         // global_addr low
  g0[3] = (unsigned)((ga >> 32) & 0x1FFFFFFu)   // global_addr high (57b total)
        | (2u << 30);                           // type = 2 ("image")
  v8i g1;
  g1[0] = (int)((2u << 16)    // data_size = 4B
              | (1u << 20)    // pad_enable
              | (7u << 22)    // pad_interval: 256 dwords
              | (3u << 25));  // pad_amount: 4 dwords
  g1[1] = (int)(256u << 16);  // tensor_dim0 = 256 (low 16 bits)
  g1[2] = (int)(64u  << 16);  // tensor_dim1 = 64 (low 16 bits)
  g1[3] = (int)(256u << 16);  // tensor_dim1 hi = 0, tile_dim0 = 256
  g1[4] = 64;                 // tile_dim1 = 64, tile_dim2 = 0
  g1[5] = 256;                // tensor_dim0_stride = 256 (low 32)
  g1[6] = 0;                  // stride0 hi, tensor_dim1_stride lo
  g1[7] = 0;
  v4i z4 = {0, 0, 0, 0};
#if __clang_major__ >= 23
  v8i z8 = {0, 0, 0, 0, 0, 0, 0, 0};
  __builtin_amdgcn_tensor_load_to_lds(g0, g1, z4, z4, z8, 0);
#else
  __builtin_amdgcn_tensor_load_to_lds(g0, g1, z4, z4, 0);
#endif
}
#endif // HAVE_TDM

extern __shared__ float smem[];

__global__ __launch_bounds__(256) void attnpool_main(
    const float* __restrict__ x, const float* __restrict__ W,
    const float* __restrict__ bvec, float* __restrict__ out,
    float* __restrict__ s_ws) {
  // LDS layout (all indexing kept as integer offsets into `smem` so the
  // compiler keeps addrspace(3) and emits ds_load/ds_store):
  //   [0,                TT*ROWF)      tile buffer 0
  //   [TT*ROWF,        2*TT*ROWF)      tile buffer 1
  //   [2*TT*ROWF,      2*TT*ROWF+TT)   w buffer 0
  //   [2*TT*ROWF+TT,   2*TT*ROWF+2*TT) w buffer 1
  const int tid  = threadIdx.x;
  const int lane = tid & 31;
  const int wave = tid >> 5;
  const int b     = blockIdx.x >> 3;
  const int chunk = blockIdx.x & 7;
  const int t0    = chunk * TCHUNK;
  const float* xb = x + ((size_t)b * TLEN + t0) * DDIM;

  // Per-lane slice of W: lane l owns d = 8l .. 8l+7 (kept in VGPRs all kernel)
  const float4 w0 = *(const float4*)(W + lane * 8);
  const float4 w1 = *(const float4*)(W + lane * 8 + 4);

  v8f acc0 = {0.f, 0.f, 0.f, 0.f, 0.f, 0.f, 0.f, 0.f};
  v8f acc1 = {0.f, 0.f, 0.f, 0.f, 0.f, 0.f, 0.f, 0.f};
  float s_acc = 0.f;

#ifdef HAVE_TDM
  if (wave == 0)
    tdm_load_tile(xb, (unsigned)(size_t)(void*)&smem[0]);
#endif

  for (int tile = 0; tile < TILES; ++tile) {
    const int p    = tile & 1;
    const int xoff = p * (TT * ROWF);          // tile buffer base (floats)
    const int woff = 2 * TT * ROWF + p * TT;   // w buffer base (floats)
#ifdef HAVE_TDM
    if (wave == 0) {
      if (tile + 1 < TILES) {
        tdm_load_tile(xb + (size_t)(tile + 1) * TT * DDIM,
                      (unsigned)(size_t)(void*)&smem[(p ^ 1) * (TT * ROWF)]);
        __builtin_amdgcn_s_wait_tensorcnt((short)1);  // current tile landed
      } else {
        __builtin_amdgcn_s_wait_tensorcnt((short)0);
      }
    }
#else
    // Fallback: cooperative synchronous copy (float4 granularity)
    {
      const float* gsrc = xb + (size_t)tile * TT * DDIM;
      for (int j = tid; j < TT * 64; j += 256) {
        int r = j >> 6, c = j & 63;
        *(float4*)&smem[xoff + r * ROWF + c * 4] =
            *(const float4*)(gsrc + r * DDIM + c * 4);
      }
    }
#endif
    __syncthreads();  // tile visible to all waves

    // ---- Phase A: scores w[t] = exp(tanh(x[t,:].W + b[t])) ----
    {
      const int tl = wave * 8;  // 8 timesteps per wave
      for (int i = 0; i < 8; ++i) {
        const int rbase = xoff + (tl + i) * ROWF + lane * 8;
        float4 a = *(const float4*)&smem[rbase];
        float4 c = *(const float4*)&smem[rbase + 4];
        float dsum = a.x * w0.x + a.y * w0.y + a.z * w0.z + a.w * w0.w
                   + c.x * w1.x + c.y * w1.y + c.z * w1.z + c.w * w1.w;
        for (int off = 16; off; off >>= 1)
          dsum += __shfl_xor(dsum, off, 32);
        float wt = fast_exp(fast_tanh(dsum + bvec[t0 + tile * TT + tl + i]));
        if (lane == 0) {
          smem[woff + tl + i] = wt;
          s_acc += wt;
        }
      }
    }
    __syncthreads();  // w buffer ready

    // ---- Phase B: WMMA pooling  acc[d] += sum_t w[t] * x[t,d] ----
    // A (16x4): A[m][k] = w[tb+k] for all m (broadcast rows).
    //   layout: lanes 0-15 -> v0=K0, v1=K1 ; lanes 16-31 -> v0=K2, v1=K3
    // B (4x16): B[k][n] = x[tb+k][d0+n]
    //   layout: v0 = rows K0/K2 (half-wave split), v1 = rows K1/K3
    {
      const int half2 = (lane >> 4) * 2;  // 0 or 2
      const int nl    = lane & 15;
      const int d0    = wave * 16;
      const int d1    = (wave + 8) * 16;
      for (int kk = 0; kk < 16; ++kk) {
        const int tb = kk * 4;
        const int r0 = xoff + (tb + half2) * ROWF;      // row K0/K2
        const int r1 = r0 + ROWF;                       // row K1/K3
        v2f A;
        A[0] = smem[woff + tb + half2];
        A[1] = smem[woff + tb + half2 + 1];
        v2f B0;
        B0[0] = smem[r0 + d0 + nl];
        B0[1] = smem[r1 + d0 + nl];
        v2f B1;
        B1[0] = smem[r0 + d1 + nl];
        B1[1] = smem[r1 + d1 + nl];
        acc0 = __builtin_amdgcn_wmma_f32_16x16x4_f32(
            false, A, false, B0, (short)0, acc0, false, false);
        acc1 = __builtin_amdgcn_wmma_f32_16x16x4_f32(
            false, A, false, B1, (short)0, acc1, false, false);
      }
    }
    __syncthreads();  // all reads of buffer p done before it is overwritten
  }

  // All 16 C rows are identical (A rows identical); extract row M=0:
  // C vgpr0, lanes 0-15, N = lane.
  if (lane < 16) {
    atomicAdd(out + b * DDIM + wave * 16 + lane, acc0[0]);
    atomicAdd(out + b * DDIM + (wave + 8) * 16 + lane, acc1[0]);
  }
  if (lane == 0)
    atomicAdd(s_ws + b, s_acc);
}

__global__ __launch_bounds__(256) void attnpool_finalize(
    float* __restrict__ out, const float* __restrict__ s_ws) {
  const int b = blockIdx.x;
  const int d = threadIdx.x;
  out[b * DDIM + d] = out[b * DDIM + d] / (s_ws[b] + EPS_F);
}

extern "C" void kernel_launch(void* const* d_in, const int* in_sizes, int n_in,
                              void* d_out, int out_size, void* d_ws, size_t ws_size,
                              hipStream_t stream) {
  (void)in_sizes; (void)n_in; (void)out_size; (void)ws_size;
  const float* x    = (const float*)d_in[0];
  const float* W    = (const float*)d_in[1];
  const float* bvec = (const float*)d_in[2];
  float* out  = (float*)d_out;
  float* s_ws = (float*)d_ws;

  // Zero accumulators every call (atomic accumulation, graph-replay safe).
  hipMemsetAsync(d_out, 0, (size_t)BATCH * DDIM * sizeof(float), stream);
  hipMemsetAsync(d_ws, 0, (size_t)BATCH * sizeof(float), stream);

  const size_t smem_bytes = (size_t)(2 * TT * ROWF + 2 * TT) * sizeof(float);
  attnpool_main<<<BATCH * 8, 256, smem_bytes, stream>>>(x, W, bvec, out, s_ws);
  attnpool_finalize<<<BATCH, 256, 0, stream>>>(out, s_ws);
}